// EdgeExistanceModule_54228257080068
// MI455X (gfx1250) — compile-verified
//
#include <hip/hip_runtime.h>
#include <hip/hip_bf16.h>

typedef __attribute__((ext_vector_type(2))) float v2f;
typedef __attribute__((ext_vector_type(4))) float f4;
typedef __attribute__((ext_vector_type(8))) float v8f;

#define NN 2048
#define DD 16

// ---------------------------------------------------------------------------
// Kernel A: per-node partial dots via V_WMMA_F32_16X16X4_F32.
//   a[i] = dot(e[i], W[0:16]),  c[i] = dot(e[i], W[16:32])
// One wave (32 threads) per 16-node tile; 4 WMMA k-steps of 16x16x4.
// A layout (32-bit A 16x4): lanes 0-15 -> M=lane, {K, K+1}; lanes 16-31 -> M,
// {K+2, K+3}. B (4x16) mirrored: VGPR0={K,K+2}, VGPR1={K+1,K+3}, N=lane&15.
// C layout: VGPR r -> M=r (lanes 0-15, N=lane) / M=r+8 (lanes 16-31, N=lane-16).
// Only columns N=0 (a) and N=1 (c) of B/C are used; the rest are zero.
// ---------------------------------------------------------------------------
__global__ __launch_bounds__(32)
void node_dots_wmma(const float* __restrict__ emb, const float* __restrict__ W,
                    float* __restrict__ a_out, float* __restrict__ c_out) {
    const int lane  = threadIdx.x;
    const int m     = lane & 15;          // M (for A) / N (for B)
    const int upper = lane >> 4;          // 0: K,K+1   1: K+2,K+3
    const int row   = blockIdx.x * 16 + m;

    v8f acc = {};
#pragma unroll
    for (int s = 0; s < 4; ++s) {
        const int kk = s * 4 + upper * 2;
        v2f a;
        a.x = emb[row * DD + kk];
        a.y = emb[row * DD + kk + 1];
        // B[k][n]: n==0 -> W[k] (first half), n==1 -> W[16+k] (second half)
        const float w0  = W[kk],      w1  = W[kk + 1];
        const float w16 = W[16 + kk], w17 = W[16 + kk + 1];
        v2f b;
        b.x = (m == 0) ? w0 : ((m == 1) ? w16 : 0.0f);   // branchless: v_cndmask
        b.y = (m == 0) ? w1 : ((m == 1) ? w17 : 0.0f);
        acc = __builtin_amdgcn_wmma_f32_16x16x4_f32(
            /*neg_a=*/false, a, /*neg_b=*/false, b,
            /*c_mod=*/(short)0, acc, /*reuse_a=*/false, /*reuse_b=*/false);
    }

    // Column N=0 -> a[], column N=1 -> c[]. Lanes {0,16} hold N=0, {1,17} N=1.
    const int mbase = blockIdx.x * 16 + upper * 8;
    if (m == 0) {
#pragma unroll
        for (int r = 0; r < 8; ++r) a_out[mbase + r] = acc[r];
    } else if (m == 1) {
#pragma unroll
        for (int r = 0; r < 8; ++r) c_out[mbase + r] = acc[r];
    }
}

// ---------------------------------------------------------------------------
// Kernel B: stream edge_embeddings[p] = concat(e[i], e[j]), p = i*N + j.
// Unit of work = one float4 quad (16B). 8 quads per row: quads 0-3 from e[i],
// 4-7 from e[j]. Consecutive lanes -> consecutive 16B of output (coalesced);
// non-temporal b128 stores for the 512 MiB stream.
// ---------------------------------------------------------------------------
__global__ __launch_bounds__(256)
void write_edge_emb(const f4* __restrict__ emb4, f4* __restrict__ out4,
                    unsigned total) {
    unsigned idx    = blockIdx.x * blockDim.x + threadIdx.x;
    unsigned stride = gridDim.x * blockDim.x;
    for (unsigned q = idx; q < total; q += stride) {
        const unsigned p = q >> 3;        // pair index
        const unsigned k = q & 7;         // quad within 32-float row
        const unsigned i = p >> 11;       // p / 2048
        const unsigned j = p & 2047;
        const unsigned srcq = (k < 4) ? (i * 4 + k) : (j * 4 + (k - 4));
        const f4 v = emb4[srcq];
        __builtin_nontemporal_store(v, &out4[q]);
    }
}

// ---------------------------------------------------------------------------
// Kernel C: logits[p] = (i!=j) ? a[i]+c[j]+b : -10 ; true_edges[p] = 0.
// Each thread handles 4 consecutive pairs (same i, j..j+3) -> float4 I/O.
// ---------------------------------------------------------------------------
__global__ __launch_bounds__(256)
void logits_and_zero(const float* __restrict__ a, const f4* __restrict__ c4,
                     const float* __restrict__ b_ptr,
                     f4* __restrict__ logits4, f4* __restrict__ true4,
                     unsigned total4) {
    unsigned t = blockIdx.x * blockDim.x + threadIdx.x;
    if (t >= total4) return;
    const unsigned i  = t >> 9;           // (4t) / 2048
    const unsigned j0 = (t & 511) * 4;
    const float ai = a[i];
    const float bb = b_ptr[0];
    const f4 cj = c4[t & 511];
    f4 lg;
    lg.x = (j0 + 0 != i) ? (ai + cj.x + bb) : -10.0f;
    lg.y = (j0 + 1 != i) ? (ai + cj.y + bb) : -10.0f;
    lg.z = (j0 + 2 != i) ? (ai + cj.z + bb) : -10.0f;
    lg.w = (j0 + 3 != i) ? (ai + cj.w + bb) : -10.0f;
    __builtin_nontemporal_store(lg, &logits4[t]);
    const f4 z = {0.0f, 0.0f, 0.0f, 0.0f};
    __builtin_nontemporal_store(z, &true4[t]);
}

// ---------------------------------------------------------------------------
// Kernel D: scatter 1.0 at src*N + dst (runs after zero-fill via stream order).
// ---------------------------------------------------------------------------
__global__ __launch_bounds__(256)
void scatter_ones(const int* __restrict__ edges, float* __restrict__ true_edges,
                  int E) {
    int t = blockIdx.x * blockDim.x + threadIdx.x;
    if (t >= E) return;
    const int src = edges[t];
    const int dst = edges[E + t];
    true_edges[src * NN + dst] = 1.0f;
}

extern "C" void kernel_launch(void* const* d_in, const int* in_sizes, int n_in,
                              void* d_out, int out_size, void* d_ws, size_t ws_size,
                              hipStream_t stream) {
    const float* emb   = (const float*)d_in[0];   // [2048, 16] f32
    const int*   edges = (const int*)d_in[1];     // [2, 65536] i32
    const float* W     = (const float*)d_in[2];   // [1, 32] f32
    const float* b     = (const float*)d_in[3];   // [1] f32

    const long long nn = (long long)NN * NN;
    float* out_emb    = (float*)d_out;            // [nn, 32]
    float* out_logits = out_emb + nn * 2 * DD;    // [nn]
    float* out_true   = out_logits + nn;          // [nn]

    float* a_ws = (float*)d_ws;                   // [2048]
    float* c_ws = a_ws + NN;                      // [2048]

    // A: WMMA GEMV  (128 one-wave tiles)
    node_dots_wmma<<<NN / 16, 32, 0, stream>>>(emb, W, a_ws, c_ws);

    // B: 512 MiB embedding stream (grid-stride, 16 quads/thread)
    const unsigned totalq = (unsigned)(nn * 8);   // float4 units
    write_edge_emb<<<8192, 256, 0, stream>>>((const f4*)emb, (f4*)out_emb, totalq);

    // C: logits + zero-fill true_edges (1M threads, 4 pairs each)
    const unsigned total4 = (unsigned)(nn / 4);
    logits_and_zero<<<total4 / 256, 256, 0, stream>>>(
        a_ws, (const f4*)c_ws, b, (f4*)out_logits, (f4*)out_true, total4);

    // D: scatter true edges
    const int E = in_sizes[1] / 2;
    scatter_ones<<<(E + 255) / 256, 256, 0, stream>>>(edges, out_true, E);
}